// PrimaryCaps_76510547411539
// MI455X (gfx1250) — compile-verified
//
#include <hip/hip_runtime.h>
#include <hip/hip_bf16.h>

typedef __attribute__((ext_vector_type(2))) float v2f;
typedef __attribute__((ext_vector_type(8))) float v8f;

#define K_A        512      // contraction (input channels)
#define N_IMG      64
#define HW         1024     // 32*32 pixels per image
#define O_POSE     512      // pose output channels (32 caps * 16)
#define O_ACT      32       // activation output channels
#define O_TILES    34       // ceil(544/16)
#define PIX_PER_BLK 32      // pixels (GEMM-N) per block: two 16-wide WMMA tiles
#define LDS_STRIDE 518      // dwords per pixel row: even (8B-aligned b64) and
                            // 518 % 64 == 6 -> conflict-free for 32 lanes

__device__ __forceinline__ float sigmoidf(float v) {
    return 1.0f / (1.0f + __expf(-v));
}

extern "C" __global__ __launch_bounds__(256)
void primarycaps_wmma_f32(const float* __restrict__ x,
                          const float* __restrict__ Wp,
                          const float* __restrict__ bp,
                          const float* __restrict__ Wa,
                          const float* __restrict__ ba,
                          float* __restrict__ out)
{
    extern __shared__ float lds[];   // [PIX_PER_BLK][LDS_STRIDE], x-panel transposed

    const int tid  = threadIdx.x;
    const int lane = tid & 31;
    const int wave = tid >> 5;
    const int bid  = blockIdx.x;
    const int n    = bid >> 5;       // image index
    const int p0   = (bid & 31) * PIX_PER_BLK;  // pixel base within image

    // ---- Stage x[n][:, p0:p0+32] into LDS as [pixel][k] ------------------
    // Per wave-instruction: 32 consecutive floats from global (128B coalesced),
    // LDS store lane-stride 518 dwords -> conflict-free.
    {
        const int p  = tid & 31;
        const int k0 = tid >> 5;     // 0..7
        const float* gx = x + (size_t)n * K_A * HW + p0 + p;
        float* lrow = lds + p * LDS_STRIDE;
        #pragma unroll 8
        for (int k = k0; k < K_A; k += 8)
            lrow[k] = gx[(size_t)k * HW];
    }
    __syncthreads();

    // ---- Per-lane fragment geometry (ISA 7.12.2 layouts) -----------------
    const int lm = lane & 15;            // M (A) / N (B) index within tile
    const int ko = (lane >> 4) << 1;     // K sub-offset: 0 or 2
    const float* lb0 = lds + (size_t)(lm)      * LDS_STRIDE + ko;  // B frag, px 0-15
    const float* lb1 = lds + (size_t)(lm + 16) * LDS_STRIDE + ko;  // B frag, px 16-31

    // ---- Output-channel tiles round-robined over the 8 waves -------------
    for (int ot = wave; ot < O_TILES; ot += 8) {
        const int  o0     = ot * 16;
        const bool is_act = (o0 >= O_POSE);
        const float* Wbase = is_act ? (Wa + (size_t)(o0 - O_POSE) * K_A)
                                    : (Wp + (size_t)o0 * K_A);
        // A fragment: lane L holds W[o0 + (L&15)][k + 2*(L>>4)] (+1)
        const float* ap = Wbase + (size_t)lm * K_A + ko;

        v8f acc0 = {0.f, 0.f, 0.f, 0.f, 0.f, 0.f, 0.f, 0.f};
        v8f acc1 = {0.f, 0.f, 0.f, 0.f, 0.f, 0.f, 0.f, 0.f};

        #pragma unroll 4
        for (int k = 0; k < K_A; k += 4) {
            v2f a  = *(const v2f*)(ap  + k);   // global_load_b64 (L2-resident W)
            v2f b0 = *(const v2f*)(lb0 + k);   // ds_load_b64
            v2f b1 = *(const v2f*)(lb1 + k);   // ds_load_b64
            acc0 = __builtin_amdgcn_wmma_f32_16x16x4_f32(
                       false, a, false, b0, (short)0, acc0, false, false);
            acc1 = __builtin_amdgcn_wmma_f32_16x16x4_f32(
                       false, a, false, b1, (short)0, acc1, false, false);
        }

        // ---- Epilogue: C/D layout -> VGPR r holds row (r + 8*(lane>=16)) --
        const int rhi = (lane >> 4) << 3;    // 0 or 8

        if (!is_act) {
            const size_t base = ((size_t)n * O_POSE + o0) * HW + p0;
            #pragma unroll
            for (int r = 0; r < 8; ++r) {
                const int   m    = r + rhi;
                const float bias = bp[o0 + m];
                out[base + (size_t)m * HW + lm]      = acc0[r] + bias;
                out[base + (size_t)m * HW + 16 + lm] = acc1[r] + bias;
            }
        } else {
            const size_t abase = (size_t)N_IMG * O_POSE * HW
                               + ((size_t)n * O_ACT + (o0 - O_POSE)) * HW + p0;
            #pragma unroll
            for (int r = 0; r < 8; ++r) {
                const int   m    = r + rhi;
                const float bias = ba[o0 - O_POSE + m];
                out[abase + (size_t)m * HW + lm]      = sigmoidf(acc0[r] + bias);
                out[abase + (size_t)m * HW + 16 + lm] = sigmoidf(acc1[r] + bias);
            }
        }
    }
}

extern "C" void kernel_launch(void* const* d_in, const int* in_sizes, int n_in,
                              void* d_out, int out_size, void* d_ws, size_t ws_size,
                              hipStream_t stream) {
    const float* x  = (const float*)d_in[0];   // [64, 512, 32, 32]
    const float* Wp = (const float*)d_in[1];   // [512, 512]
    const float* bp = (const float*)d_in[2];   // [512]
    const float* Wa = (const float*)d_in[3];   // [32, 512]
    const float* ba = (const float*)d_in[4];   // [32]
    float* out = (float*)d_out;                // poses ++ activations

    const dim3 grid(N_IMG * (HW / PIX_PER_BLK));   // 2048 blocks
    const dim3 block(256);                         // 8 wave32s
    const size_t shmem = (size_t)PIX_PER_BLK * LDS_STRIDE * sizeof(float); // ~66 KB

    hipLaunchKernelGGL(primarycaps_wmma_f32, grid, block, shmem, stream,
                       x, Wp, bp, Wa, ba, out);
}